// TripletConv_33397665694633
// MI455X (gfx1250) — compile-verified
//
#include <hip/hip_runtime.h>

// ---------------------------------------------------------------------------
// TripletConv on gfx1250 (MI455X).  B=4, N=96, DIM=128, SUB=8.
//
// e_ik = path_emb @ (We@W4) + be@W4 (etc.), ti = node_emb @ (Wn@W1) + bn@W1
// (etc.) -> precompute combined 128x8 weights, one WMMA GEMM over path_emb
// (36864x128 @ 128x24, padded to 32), one over node_emb (384x128), VALU
// max-plus triplet reduction (not a matmul), WMMA GEMM for the 8->128 output.
//
// Data movement: intermediate E is stored PLANAR (three M x 8 slabs) so the
// triplet kernel's k-streamed reads are fully coalesced (lane j reads 32
// contiguous bytes; E slabs total 3.5MB and live in the 192MB L2).  The
// per-(b,i) X row and T_k slab are staged into LDS with
// GLOBAL_LOAD_ASYNC_TO_LDS_B128 + s_wait_asynccnt.
// f32 WMMA (V_WMMA_F32_16X16X4_F32) keeps exact f32 math: the problem is
// bandwidth-bound (~38MB @ 23.3TB/s), so low-precision WMMA buys nothing.
// ---------------------------------------------------------------------------

typedef float v2f __attribute__((ext_vector_type(2)));
typedef float v8f __attribute__((ext_vector_type(8)));
typedef int   v4i __attribute__((ext_vector_type(4)));

typedef __attribute__((address_space(1))) v4i* as1_v4i;
typedef __attribute__((address_space(3))) v4i* as3_v4i;

#define BB   4
#define NN   96
#define DIMD 128
#define SUBS 8
#define ME   (BB * NN * NN)   // 36864 rows of E / Tij / out
#define MT   (BB * NN)        // 384 rows of T

// workspace layout (in floats)
#define OFF_WCT  0                          // 32 x 128 combined edge weights (transposed)
#define OFF_NCT  (OFF_WCT + 32 * 128)       // 32 x 128 combined node weights (transposed)
#define OFF_EB   (OFF_NCT + 32 * 128)       // 32  folded edge biases (+bt in ij slot)
#define OFF_TB   (OFF_EB + 32)              // 32  folded node biases
#define OFF_WOT  (OFF_TB + 32)              // 128 x 8 Wo transposed
#define OFF_EIK  (OFF_WOT + 128 * 8)        // ME x 8 planar
#define OFF_EKJ  (OFF_EIK + ME * 8)         // ME x 8 planar
#define OFF_EIJ  (OFF_EKJ + ME * 8)         // ME x 8 planar
#define OFF_TI   (OFF_EIJ + ME * 8)         // MT x 8 planar
#define OFF_TJ   (OFF_TI + MT * 8)          // MT x 8 planar
#define OFF_TK   (OFF_TJ + MT * 8)          // MT x 8 planar
#define OFF_TIJ  (OFF_TK + MT * 8)          // ME x 8 max-over-k result
// total = 1,198,144 floats ~= 4.6 MB of d_ws

#define HAVE_ASYNC_LDS                                                \
    (__has_builtin(__builtin_amdgcn_global_load_async_to_lds_b128) && \
     __has_builtin(__builtin_amdgcn_s_wait_asynccnt))

#if HAVE_ASYNC_LDS
// 16B async global->LDS copy.  Builtin prototype (from hipcc diagnostics):
//   void (v4i addrspace(1)*, v4i addrspace(3)*, imm offset, imm cpol)
// reinterpret across address spaces via integer round-trip (addrspacecast:
// AS1 keeps the 64-bit address; AS3 is the low 32 bits of the LDS aperture).
__device__ __forceinline__ void async_ld16(const void* g, void* l) {
    __builtin_amdgcn_global_load_async_to_lds_b128(
        (as1_v4i)(unsigned long long)g, (as3_v4i)(unsigned long long)l, 0, 0);
}
#endif

// ---------------------------------------------------------------------------
// Kernel 1: fold the six 8x8 triplet blocks into the 128-d projections.
//   Wt rows: [0:8)=W1(ti) [8:16)=W2(tj) [16:24)=W3(tk)
//            [24:32)=W4(e_ik) [32:40)=W5(e_kj) [40:48)=W6(e_ij)
// ---------------------------------------------------------------------------
__global__ void prep_kernel(const float* __restrict__ Wn, const float* __restrict__ bn,
                            const float* __restrict__ We, const float* __restrict__ be,
                            const float* __restrict__ Wt, const float* __restrict__ bt,
                            const float* __restrict__ Wo, float* __restrict__ ws) {
    float* WcT = ws + OFF_WCT;
    float* NcT = ws + OFF_NCT;
    float* Eb  = ws + OFF_EB;
    float* Tb  = ws + OFF_TB;
    float* WoT = ws + OFF_WOT;
    const int t = threadIdx.x;

    for (int idx = t; idx < 32 * 128; idx += blockDim.x) {
        const int n = idx / 128, k = idx % 128;
        float we_acc = 0.f, wn_acc = 0.f;
        if (n < 24) {
            const int slot = n >> 3, s = n & 7;
            #pragma unroll
            for (int d = 0; d < 8; ++d) {
                we_acc += We[k * 8 + d] * Wt[(24 + slot * 8 + d) * 8 + s];
                wn_acc += Wn[k * 8 + d] * Wt[(slot * 8 + d) * 8 + s];
            }
        }
        WcT[n * 128 + k] = we_acc;
        NcT[n * 128 + k] = wn_acc;
    }
    for (int n = t; n < 32; n += blockDim.x) {
        float eb = 0.f, tb = 0.f;
        if (n < 24) {
            const int slot = n >> 3, s = n & 7;
            #pragma unroll
            for (int d = 0; d < 8; ++d) {
                eb += be[d] * Wt[(24 + slot * 8 + d) * 8 + s];
                tb += bn[d] * Wt[(slot * 8 + d) * 8 + s];
            }
            if (slot == 2) eb += bt[s];
        }
        Eb[n] = eb;
        Tb[n] = tb;
    }
    for (int idx = t; idx < 128 * 8; idx += blockDim.x) {
        const int n = idx / 8, k = idx % 8;
        WoT[n * 8 + k] = Wo[k * 128 + n];
    }
}

// ---------------------------------------------------------------------------
// Kernel 2: X[M x 128] @ W[128 x 32] + bias via f32 WMMA, one wave per
// 16-row M-tile, results scattered into three planar M x 8 slabs:
//   n-tile0 cols 0..7 -> dst0, cols 8..15 -> dst1; n-tile1 cols 16..23 -> dst2.
// A layout (16x4 f32): lanes 0-15 = rows; v0/v1 hold K = 2*(lane>=16)+{0,1}.
// ---------------------------------------------------------------------------
__global__ void proj_wmma(const float* __restrict__ X, const float* __restrict__ WT,
                          const float* __restrict__ bias,
                          float* __restrict__ dst0, float* __restrict__ dst1,
                          float* __restrict__ dst2) {
    const int lane  = threadIdx.x;
    const int mt    = blockIdx.x;
    const int nlo   = lane & 15;
    const int khalf = (lane >> 4) * 2;

    const float* arow = X + (mt * 16 + nlo) * 128 + khalf;
    const float* b0p  = WT + nlo * 128 + khalf;
    const float* b1p  = WT + (nlo + 16) * 128 + khalf;

    v8f c0 = {};
    v8f c1 = {};
    #pragma unroll 4
    for (int kt = 0; kt < 32; ++kt) {
        v2f a  = *(const v2f*)(arow + kt * 4);
        v2f b0 = *(const v2f*)(b0p + kt * 4);
        v2f b1 = *(const v2f*)(b1p + kt * 4);
        c0 = __builtin_amdgcn_wmma_f32_16x16x4_f32(false, a, false, b0, (short)0, c0, false, false);
        c1 = __builtin_amdgcn_wmma_f32_16x16x4_f32(false, a, false, b1, (short)0, c1, false, false);
    }
    const float bs0 = bias[nlo];
    const float bs1 = bias[nlo + 16];
    const int mbase = mt * 16 + ((lane >> 4) << 3);  // C/D: VGPR r -> row r / r+8

    // c0: cols 0..15 -> slabs 0/1
    float* p0 = (nlo < 8) ? (dst0 + nlo) : (dst1 + (nlo - 8));
    #pragma unroll
    for (int r = 0; r < 8; ++r) p0[(mbase + r) * 8] = c0[r] + bs0;
    // c1: cols 16..23 -> slab 2 (cols 24..31 are padding: not stored)
    if (nlo < 8) {
        #pragma unroll
        for (int r = 0; r < 8; ++r) dst2[(mbase + r) * 8 + nlo] = c1[r] + bs1;
    }
}

// ---------------------------------------------------------------------------
// Kernel 3: triplet max-plus reduction.  Block = (b, i), 96 threads (3 full
// waves), thread = j.  X row (E_ik[b,i,*] and T_k[b,*], 3KB each) staged to
// LDS via async global->LDS b128 copies, combined in place, then:
//   t_ij[s] = max(0, max_k (X[k][s] + Y[k][s] + Z[s]))
// Y reads are fully coalesced thanks to the planar E_kj slab (L2 resident).
// ---------------------------------------------------------------------------
__global__ void triplet_kernel(const float* __restrict__ Eik, const float* __restrict__ Ekj,
                               const float* __restrict__ Eij,
                               const float* __restrict__ Ti, const float* __restrict__ Tj,
                               const float* __restrict__ Tk,
                               float* __restrict__ Tij) {
    __shared__ float Xe[NN * 8];
    __shared__ float Xt[NN * 8];
    const int b   = blockIdx.x / NN;
    const int i   = blockIdx.x % NN;
    const int tid = threadIdx.x;  // 0..95 == k (staging) == j (compute)

    const float* esrc = Eik + ((b * NN + i) * NN) * 8 + tid * 8;  // 32B per thread
    const float* tsrc = Tk + (b * NN) * 8 + tid * 8;              // 32B per thread
#if HAVE_ASYNC_LDS
    async_ld16(esrc,     &Xe[tid * 8]);
    async_ld16(esrc + 4, &Xe[tid * 8 + 4]);
    async_ld16(tsrc,     &Xt[tid * 8]);
    async_ld16(tsrc + 4, &Xt[tid * 8 + 4]);
    __builtin_amdgcn_s_wait_asynccnt(0);   // per-wave; barrier below makes it WG-wide
    __syncthreads();
    #pragma unroll
    for (int s = 0; s < 8; ++s) Xe[tid * 8 + s] += Xt[tid * 8 + s];  // X = E_ik + T_k
    __syncthreads();
#else
    #pragma unroll
    for (int s = 0; s < 8; ++s) Xe[tid * 8 + s] = esrc[s] + tsrc[s];
    __syncthreads();
#endif

    const int j = tid;
    float z[8];
    {
        const float* eij = Eij + ((b * NN + i) * NN + j) * 8;  // includes bt (folded)
        const float* tip = Ti + (b * NN + i) * 8;
        const float* tjp = Tj + (b * NN + j) * 8;
        #pragma unroll
        for (int s = 0; s < 8; ++s) z[s] = eij[s] + tip[s] + tjp[s];
    }

    float m[8];
    #pragma unroll
    for (int s = 0; s < 8; ++s) m[s] = 0.0f;  // relu(max_k v) == max(0, max_k v)

    const float* ybase = Ekj + (b * NN * NN + j) * 8;  // k = 0; stride NN*8 per k
    const int    ystr  = NN * 8;
    for (int k = 0; k < NN; ++k) {
        const float* y = ybase + k * ystr;
        __builtin_prefetch(y + 8 * ystr, 0, 1);  // global_prefetch ahead (L2 hit path)
        #pragma unroll
        for (int s = 0; s < 8; ++s) {
            const float v = Xe[k * 8 + s] + y[s] + z[s];
            m[s] = fmaxf(m[s], v);
        }
    }

    float* o = Tij + ((b * NN + i) * NN + j) * 8;
    #pragma unroll
    for (int s = 0; s < 8; ++s) o[s] = m[s];
}

// ---------------------------------------------------------------------------
// Kernel 4: out = relu(Tij[M x 8] @ Wo[8 x 128] + bo) via f32 WMMA.
// One wave per 16-row M-tile; K=8 -> 2 WMMAs per 16-col n-tile, 8 n-tiles.
// ---------------------------------------------------------------------------
__global__ void out_wmma(const float* __restrict__ Tij, const float* __restrict__ WoT,
                         const float* __restrict__ bo, float* __restrict__ out) {
    const int lane  = threadIdx.x;
    const int mt    = blockIdx.x;
    const int nlo   = lane & 15;
    const int khalf = (lane >> 4) * 2;

    const float* ar = Tij + (mt * 16 + nlo) * 8 + khalf;
    const v2f a0 = *(const v2f*)(ar);       // K 0..3 slice
    const v2f a1 = *(const v2f*)(ar + 4);   // K 4..7 slice
    const int mbase = mt * 16 + ((lane >> 4) << 3);

    #pragma unroll
    for (int nt = 0; nt < 8; ++nt) {
        const float* bp = WoT + (nt * 16 + nlo) * 8 + khalf;
        const v2f b0 = *(const v2f*)(bp);
        const v2f b1 = *(const v2f*)(bp + 4);
        v8f c = {};
        c = __builtin_amdgcn_wmma_f32_16x16x4_f32(false, a0, false, b0, (short)0, c, false, false);
        c = __builtin_amdgcn_wmma_f32_16x16x4_f32(false, a1, false, b1, (short)0, c, false, false);
        const float bb = bo[nt * 16 + nlo];
        #pragma unroll
        for (int r = 0; r < 8; ++r) {
            out[(mbase + r) * 128 + nt * 16 + nlo] = fmaxf(c[r] + bb, 0.0f);
        }
    }
}

// ---------------------------------------------------------------------------
extern "C" void kernel_launch(void* const* d_in, const int* in_sizes, int n_in,
                              void* d_out, int out_size, void* d_ws, size_t ws_size,
                              hipStream_t stream) {
    (void)in_sizes; (void)n_in; (void)out_size; (void)ws_size;

    const float* node_emb = (const float*)d_in[0];  // (384, 128)
    const float* path_emb = (const float*)d_in[1];  // (4, 96, 96, 128)
    const float* Wn = (const float*)d_in[2];        // (128, 8)
    const float* bn = (const float*)d_in[3];        // (8,)
    const float* We = (const float*)d_in[4];        // (128, 8)
    const float* be = (const float*)d_in[5];        // (8,)
    const float* Wt = (const float*)d_in[6];        // (48, 8)
    const float* bt = (const float*)d_in[7];        // (8,)
    const float* Wo = (const float*)d_in[8];        // (8, 128)
    const float* bo = (const float*)d_in[9];        // (128,)

    float* ws  = (float*)d_ws;   // needs ~4.6 MB
    float* out = (float*)d_out;  // (4, 96, 96, 128)

    // 1) fold 8x8 triplet blocks into 128-d combined weights + biases
    prep_kernel<<<1, 256, 0, stream>>>(Wn, bn, We, be, Wt, bt, Wo, ws);

    // 2) planar E slabs from path_emb, planar T slabs from node_emb
    proj_wmma<<<ME / 16, 32, 0, stream>>>(path_emb, ws + OFF_WCT, ws + OFF_EB,
                                          ws + OFF_EIK, ws + OFF_EKJ, ws + OFF_EIJ);
    proj_wmma<<<MT / 16, 32, 0, stream>>>(node_emb, ws + OFF_NCT, ws + OFF_TB,
                                          ws + OFF_TI, ws + OFF_TJ, ws + OFF_TK);

    // 3) Tij = max_k relu(X + Y + Z)   (36864 x 8)
    triplet_kernel<<<BB * NN, NN, 0, stream>>>(ws + OFF_EIK, ws + OFF_EKJ, ws + OFF_EIJ,
                                               ws + OFF_TI, ws + OFF_TJ, ws + OFF_TK,
                                               ws + OFF_TIJ);

    // 4) out = relu(Tij @ Wo + bo)     (36864 x 128)
    out_wmma<<<ME / 16, 32, 0, stream>>>(ws + OFF_TIJ, ws + OFF_WOT, bo, out);
}